// SubcenterAAMSoftmax_71708773974113
// MI455X (gfx1250) — compile-verified
//
#include <hip/hip_runtime.h>
#include <hip/hip_bf16.h>
#include <math.h>

// ---- problem constants (from reference) ----
#define NUM_CLASSES 50000
#define KSUB        3
#define NSUB        (NUM_CLASSES * KSUB)   // 150000
#define EMBD        256
#define BATCH       1024
#define SCALE_F     32.0f

// margin = 0.25
#define COS_M 0.9689124217106447f
#define SIN_M 0.24740395925452294f
#define TH_F  (-0.9689124217106447f)      // cos(pi - m) = -cos(m)
#define MM_F  0.061850989813630735f       // sin(pi - m) * m

// ---- tiling ----
#define BT        16                      // batch rows per workgroup (WMMA M tile)
#define NBT       (BATCH / BT)            // 64 batch tiles
#define BLK_CLS   128                     // classes per inner block (384 subcols = 24 N-tiles)
#define CHUNK_CLS 1024                    // classes per workgroup
#define NCHUNK    ((NUM_CLASSES + CHUNK_CLS - 1) / CHUNK_CLS)   // 49

typedef __attribute__((ext_vector_type(2))) float v2f;
typedef __attribute__((ext_vector_type(8))) float v8f;

// ---------------------------------------------------------------------------
// Kernel 1: inv row-norm of weight. One wave32 per row (256 floats).
// ---------------------------------------------------------------------------
__global__ __launch_bounds__(256) void rownorm_kernel(const float* __restrict__ W,
                                                      float* __restrict__ invn) {
    int gw   = (blockIdx.x * 256 + threadIdx.x) >> 5;   // global wave id == row
    int lane = threadIdx.x & 31;
    if (gw >= NSUB) return;
    const float4* row = (const float4*)(W + (size_t)gw * EMBD);
    float4 a = row[lane];
    float4 b = row[lane + 32];
    float acc = a.x * a.x + a.y * a.y + a.z * a.z + a.w * a.w +
                b.x * b.x + b.y * b.y + b.z * b.z + b.w * b.w;
    #pragma unroll
    for (int off = 16; off > 0; off >>= 1) acc += __shfl_xor(acc, off, 32);
    if (lane == 0) invn[gw] = 1.0f / fmaxf(sqrtf(acc), 1e-12f);
}

// ---------------------------------------------------------------------------
// Kernel 2: fused GEMM (f32 WMMA) + subcenter max + AAM margin + streaming
// logsumexp partials.  grid.x = NBT * NCHUNK, block = 256 threads (8 waves).
// Each wave keeps 3 independent accumulator tiles live, sharing one LDS
// A-fragment load per k-step (3 interleaved WMMA dependency chains).
// ---------------------------------------------------------------------------
__global__ __launch_bounds__(256, 1) void aam_main_kernel(const float* __restrict__ E,
                                                          const int*   __restrict__ labels,
                                                          const float* __restrict__ W,
                                                          const float* __restrict__ invn,
                                                          float*       __restrict__ partials) {
    __shared__ __align__(16) float eA[BT][260];          // 16B-aligned rows, conflict-free b64 reads
    __shared__ float logitsL[BT][3 * BLK_CLS + 4];       // [16][388]
    __shared__ float redM[256], redS[256], redL[256];
    __shared__ int   lab16[BT];

    const int bt   = blockIdx.x & (NBT - 1);     // batch tile
    const int ch   = blockIdx.x >> 6;            // class chunk
    const int row0 = bt * BT;

    const int t    = threadIdx.x;
    const int wave = t >> 5;
    const int lane = t & 31;
    const int hlf  = lane >> 4;                  // half-wave select (A: K pair, D: M+8)
    const int lrow = lane & 15;

    // ---- preload embedding tile into LDS (float4 stores) ----
    for (int i = t; i < BT * (EMBD / 4); i += 256) {
        int r  = i >> 6;
        int c4 = (i & 63) << 2;
        float4 v = *(const float4*)(E + (size_t)(row0 + r) * EMBD + c4);
        *(float4*)&eA[r][c4] = v;
    }
    if (t < BT) lab16[t] = labels[row0 + t];
    __syncthreads();

    // streaming-logsumexp state: thread t owns (row = t>>4, class group = t&15)
    const int myRow = t >> 4;
    const int cg    = t & 15;
    const int myLab = lab16[myRow];
    float m = -INFINITY, s = 0.0f, labAcc = 0.0f;

    const int cStart = ch * CHUNK_CLS;
    const int cEnd   = min(cStart + CHUNK_CLS, NUM_CLASSES);

    for (int cb = cStart; cb < cEnd; cb += BLK_CLS) {
        // ---- 24 N-tiles of 16 subcols; each wave owns tiles {w, w+8, w+16},
        //      all three accumulated concurrently off one shared A fragment ----
        const int tb = 3 * cb;
        int n0 = tb + (wave +  0) * 16 + lrow;
        int n1 = tb + (wave +  8) * 16 + lrow;
        int n2 = tb + (wave + 16) * 16 + lrow;
        n0 = min(n0, NSUB - 1);                  // clamp OOB (masked in accumulation)
        n1 = min(n1, NSUB - 1);
        n2 = min(n2, NSUB - 1);
        const float* wp0 = W + (size_t)n0 * EMBD;
        const float* wp1 = W + (size_t)n1 * EMBD;
        const float* wp2 = W + (size_t)n2 * EMBD;
        const float  sc0 = invn[n0];
        const float  sc1 = invn[n1];
        const float  sc2 = invn[n2];

        v8f acc0 = {}, acc1 = {}, acc2 = {};
        #pragma unroll 2
        for (int ks = 0; ks < EMBD / 4; ++ks) {
            // A frag (16x4 f32): lanes 0-15 -> K={4k,4k+1}, lanes 16-31 -> K={4k+2,4k+3}
            v2f a = *(const v2f*)&eA[lrow][4 * ks + 2 * hlf];
            // B frags (4x16 f32): each lane consumes 2 of the 4 row floats;
            // the lane-half partner consumes the other 2 -> full 16B utilization.
            float4 w0 = *(const float4*)(wp0 + 4 * ks);
            float4 w1 = *(const float4*)(wp1 + 4 * ks);
            float4 w2 = *(const float4*)(wp2 + 4 * ks);
            v2f b0, b1, b2;
            b0.x = hlf ? w0.y : w0.x;  b0.y = hlf ? w0.w : w0.z;
            b1.x = hlf ? w1.y : w1.x;  b1.y = hlf ? w1.w : w1.z;
            b2.x = hlf ? w2.y : w2.x;  b2.y = hlf ? w2.w : w2.z;
            acc0 = __builtin_amdgcn_wmma_f32_16x16x4_f32(false, a, false, b0,
                                                         (short)0, acc0, false, false);
            acc1 = __builtin_amdgcn_wmma_f32_16x16x4_f32(false, a, false, b1,
                                                         (short)0, acc1, false, false);
            acc2 = __builtin_amdgcn_wmma_f32_16x16x4_f32(false, a, false, b2,
                                                         (short)0, acc2, false, false);
        }
        // scale by per-column inv_norm and spill cosine tiles to LDS
        #pragma unroll
        for (int v = 0; v < 8; ++v) {
            const int r = v + 8 * hlf;
            logitsL[r][(wave +  0) * 16 + lrow] = acc0[v] * sc0;
            logitsL[r][(wave +  8) * 16 + lrow] = acc1[v] * sc1;
            logitsL[r][(wave + 16) * 16 + lrow] = acc2[v] * sc2;
        }
        __syncthreads();

        // ---- subcenter max + margin + streaming logsumexp over 128 classes ----
        #pragma unroll
        for (int i = 0; i < BLK_CLS / 16; ++i) {
            const int cl = cg + 16 * i;          // local class 0..127
            const int c  = cb + cl;
            if (c < cEnd) {
                float c0 = logitsL[myRow][3 * cl + 0];
                float c1 = logitsL[myRow][3 * cl + 1];
                float c2 = logitsL[myRow][3 * cl + 2];
                float cosv = fmaxf(c0, fmaxf(c1, c2));
                float x;
                if (c == myLab) {
                    float sine = sqrtf(fmaxf(0.0f, 1.0f - cosv * cosv));
                    float phi  = cosv * COS_M - sine * SIN_M;
                    phi = (cosv > TH_F) ? phi : (cosv - MM_F);
                    x = SCALE_F * phi;
                    labAcc += x;                 // label logit
                } else {
                    x = SCALE_F * cosv;
                }
                if (x > m) { s = s * expf(m - x) + 1.0f; m = x; }
                else       { s += expf(x - m); }
            }
        }
        __syncthreads();
    }

    // ---- merge the 16 partials per row, write chunk partial ----
    redM[t] = m; redS[t] = s; redL[t] = labAcc;
    __syncthreads();
    if (t < BT) {
        float M = -INFINITY, S = 0.0f, L = 0.0f;
        #pragma unroll
        for (int j = 0; j < 16; ++j) {
            float mm = redM[t * 16 + j];
            float ss = redS[t * 16 + j];
            L += redL[t * 16 + j];
            if (ss > 0.0f) {
                if (mm > M) { S = S * expf(M - mm) + ss; M = mm; }
                else        { S += ss * expf(mm - M); }
            }
        }
        float* p = partials + (((size_t)bt * NCHUNK + ch) * BT + t) * 3;
        p[0] = M; p[1] = S; p[2] = L;
    }
}

// ---------------------------------------------------------------------------
// Kernel 3: merge chunk partials per row, mean loss -> d_out[0].
// ---------------------------------------------------------------------------
__global__ __launch_bounds__(1024, 1) void aam_finish_kernel(const float* __restrict__ partials,
                                                             float* __restrict__ out) {
    __shared__ float red[1024];
    const int r  = threadIdx.x;          // batch row
    const int bt = r >> 4;
    const int rr = r & 15;
    float M = -INFINITY, S = 0.0f, L = 0.0f;
    for (int ch = 0; ch < NCHUNK; ++ch) {
        const float* p = partials + (((size_t)bt * NCHUNK + ch) * BT + rr) * 3;
        float mm = p[0], ss = p[1];
        L += p[2];
        if (ss > 0.0f) {
            if (mm > M) { S = S * expf(M - mm) + ss; M = mm; }
            else        { S += ss * expf(mm - M); }
        }
    }
    red[r] = (M + logf(S)) - L;          // per-row cross-entropy
    __syncthreads();
    #pragma unroll
    for (int off = 512; off > 0; off >>= 1) {
        if (r < off) red[r] += red[r + off];
        __syncthreads();
    }
    if (r == 0) out[0] = red[0] * (1.0f / (float)BATCH);
}

// ---------------------------------------------------------------------------
extern "C" void kernel_launch(void* const* d_in, const int* in_sizes, int n_in,
                              void* d_out, int out_size, void* d_ws, size_t ws_size,
                              hipStream_t stream) {
    const float* E      = (const float*)d_in[0];   // [1024, 256] f32
    const int*   labels = (const int*)  d_in[1];   // [1024] int
    const float* W      = (const float*)d_in[2];   // [150000, 256] f32

    float* invn     = (float*)d_ws;                // 150000 floats (pad to 150016)
    float* partials = invn + 150016;               // 64*49*16*3 = 150528 floats

    rownorm_kernel<<<NSUB / 8, 256, 0, stream>>>(W, invn);
    aam_main_kernel<<<NBT * NCHUNK, 256, 0, stream>>>(E, labels, W, invn, partials);
    aam_finish_kernel<<<1, 1024, 0, stream>>>(partials, (float*)d_out);
}